// GeneralizedGraphDiffusion_23888608100957
// MI455X (gfx1250) — compile-verified
//
#include <hip/hip_runtime.h>

typedef float v2f __attribute__((ext_vector_type(2)));
typedef float v8f __attribute__((ext_vector_type(8)));

#define SDIM 8
#define NDIM 4096
#define FDIM 128            // F_IN == F_OUT == 128
#define WAVES 8             // waves per workgroup
#define KSLICE (NDIM / WAVES)  // 512

__launch_bounds__(256, 1)
__global__ void ggd_fused_kernel(const float* __restrict__ theta,
                                 const float* __restrict__ T,
                                 const float* __restrict__ x,
                                 const float* __restrict__ adj,
                                 const float* __restrict__ Wfc,
                                 const float* __restrict__ bfc,
                                 const float* __restrict__ alpha,
                                 float* __restrict__ out)
{
    // red[wave][m (16)][n (128)]  -> 8*16*128 floats = 64 KB
    __shared__ float lds[WAVES * 16 * FDIM];

    const int tid  = threadIdx.x;
    const int wave = tid >> 5;
    const int lane = tid & 31;
    const int half = lane >> 4;   // which K-pair half of the wave
    const int lm   = lane & 15;

    const int m0 = blockIdx.x * 16;      // row block
    const int m  = m0 + lm;              // this lane's A-matrix row

    float th[SDIM];
#pragma unroll
    for (int s = 0; s < SDIM; ++s) th[s] = theta[s];   // uniform -> SGPRs

    // ---------------- Phase 1: partial (q @ x) over this wave's K slice ----
    const v8f vzero = {0.f, 0.f, 0.f, 0.f, 0.f, 0.f, 0.f, 0.f};
    v8f acc[8];
#pragma unroll
    for (int t = 0; t < 8; ++t) acc[t] = vzero;

    const size_t rowoff = (size_t)m * NDIM;
    const int kbeg = wave * KSLICE;
    const int kend = kbeg + KSLICE;

    for (int k0 = kbeg; k0 < kend; k0 += 4) {
        const int ka = k0 + 2 * half;    // this lane's first K (A layout: vgpr j -> k = 2*half + j)

        // A-tile pair: q[m][ka], q[m][ka+1]  (q = (sum_s th[s]*T[s]) * adj)
        float qx = 0.f, qy = 0.f;
#pragma unroll
        for (int s = 0; s < SDIM; ++s) {
            const float2 tv = *(const float2*)(T + (size_t)s * NDIM * NDIM + rowoff + ka);
            qx = fmaf(th[s], tv.x, qx);
            qy = fmaf(th[s], tv.y, qy);
        }
        const float2 av = *(const float2*)(adj + rowoff + ka);
        v2f A;
        A.x = qx * av.x;
        A.y = qy * av.y;

        // B-tiles: x[ka + j][t*16 + lm]  (B layout mirrors A with m<->n)
        const float* xr0 = x + (size_t)ka * FDIM;
        const float* xr1 = xr0 + FDIM;
#pragma unroll
        for (int t = 0; t < 8; ++t) {
            v2f B;
            B.x = xr0[t * 16 + lm];
            B.y = xr1[t * 16 + lm];
            acc[t] = __builtin_amdgcn_wmma_f32_16x16x4_f32(
                         false, A, false, B, (short)0, acc[t], false, false);
        }
    }

    // ------------- Phase 2: fixed-order cross-wave reduction + PReLU -------
    // C/D layout: vgpr j, lane -> (m = j + 8*half, n = t*16 + lm)
#pragma unroll
    for (int t = 0; t < 8; ++t) {
#pragma unroll
        for (int j = 0; j < 8; ++j) {
            lds[wave * (16 * FDIM) + (j + 8 * half) * FDIM + t * 16 + lm] = acc[t][j];
        }
    }
    __syncthreads();

#pragma unroll
    for (int i = 0; i < 8; ++i) {
        const int e = i * 256 + tid;                 // 0 .. 2047, exclusive per thread
        float ssum = 0.f;
#pragma unroll
        for (int w = 0; w < WAVES; ++w) ssum += lds[w * (16 * FDIM) + e];  // fixed order
        const float al = alpha[e & (FDIM - 1)];
        lds[e] = (ssum >= 0.f) ? ssum : al * ssum;   // out1 block lives in lds[0..2047]
    }
    __syncthreads();

    // ---------------- Phase 3: FC  out = out1 @ Wfc^T + b ------------------
    const int n0 = wave * 16;                        // f_out block for this wave
    v8f o = vzero;
#pragma unroll
    for (int k0 = 0; k0 < FDIM; k0 += 4) {
        const int kk = k0 + 2 * half;
        v2f A;
        A.x = lds[lm * FDIM + kk];
        A.y = lds[lm * FDIM + kk + 1];
        v2f B;                                       // B[k][n] = Wfc[n][k]
        B.x = Wfc[(size_t)(n0 + lm) * FDIM + kk];
        B.y = Wfc[(size_t)(n0 + lm) * FDIM + kk + 1];
        o = __builtin_amdgcn_wmma_f32_16x16x4_f32(
                false, A, false, B, (short)0, o, false, false);
    }

    const float bias = bfc[n0 + lm];
#pragma unroll
    for (int j = 0; j < 8; ++j) {
        out[(size_t)(m0 + j + 8 * half) * FDIM + n0 + lm] = o[j] + bias;
    }
}

extern "C" void kernel_launch(void* const* d_in, const int* in_sizes, int n_in,
                              void* d_out, int out_size, void* d_ws, size_t ws_size,
                              hipStream_t stream) {
    const float* theta = (const float*)d_in[0];
    const float* T     = (const float*)d_in[1];
    const float* x     = (const float*)d_in[2];
    const float* adj   = (const float*)d_in[3];
    const float* Wfc   = (const float*)d_in[4];
    const float* bfc   = (const float*)d_in[5];
    const float* alpha = (const float*)d_in[6];
    float* out = (float*)d_out;

    ggd_fused_kernel<<<NDIM / 16, 256, 0, stream>>>(theta, T, x, adj, Wfc, bfc, alpha, out);
}